// KAN_15333033247266
// MI455X (gfx1250) — compile-verified
//
#include <hip/hip_runtime.h>

// Problem constants (match reference)
#define B_TOT 8192
#define D_IN  64
#define HID   16
#define K_CLS 32

typedef float v2f __attribute__((ext_vector_type(2)));
typedef float v8f __attribute__((ext_vector_type(8)));

// ---------------------------------------------------------------------------
// Stage 1: inner per-edge MLPs. sk[b,k] = sum_{d,h} relu(x*w1+b1)*w2 + sum_d b2
// One thread per batch row; blockIdx.y = k. Weight indices depend only on
// (blockIdx.y, d, h) -> wave-uniform -> scalar-cache loads, VALU fma/max/fma.
// ---------------------------------------------------------------------------
__global__ __launch_bounds__(256) void kan_inner(
    const float* __restrict__ x,
    const float* __restrict__ iw1,
    const float* __restrict__ ib1,
    const float* __restrict__ iw2,
    const float* __restrict__ ib2,
    float* __restrict__ sk)
{
    const int k = blockIdx.y;
    const int b = blockIdx.x * 256 + threadIdx.x;

    const float* __restrict__ W1 = iw1 + k * (D_IN * HID);
    const float* __restrict__ B1 = ib1 + k * (D_IN * HID);
    const float* __restrict__ W2 = iw2 + k * (D_IN * HID);
    const float* __restrict__ B2 = ib2 + k * D_IN;
    const float* __restrict__ xr = x + b * D_IN;

    float acc = 0.f;
#pragma unroll 2
    for (int d = 0; d < D_IN; ++d) {
        const float xv = xr[d];
#pragma unroll
        for (int h = 0; h < HID; ++h) {
            float t = fmaf(xv, W1[d * HID + h], B1[d * HID + h]);
            t = fmaxf(t, 0.f);
            acc = fmaf(t, W2[d * HID + h], acc);
        }
    }
    // bias-2 row sum (wave-uniform)
    float s2 = 0.f;
#pragma unroll
    for (int d = 0; d < D_IN; ++d) s2 += B2[d];

    sk[b * K_CLS + k] = acc + s2;
}

// ---------------------------------------------------------------------------
// Stage 2: outer MLPs as an exact block-diagonal GEMM on WMMA f32 16x16x4.
// For a 16-batch x 16-k tile:
//   out[m,n] = sum_{c=(k',h) in 0..255} G[m,c] * Bm[c,n]
//   G[m,(k',h)]  = relu(sk[m,k']*ow1[k',h] + ob1[k',h])
//   Bm[(k',h),n] = (n==k') ? ow2[k',h] : 0
// One wave per tile. G staged in LDS so A fragments are aligned ds_load_b64.
// ---------------------------------------------------------------------------
__global__ __launch_bounds__(64) void kan_outer(
    const float* __restrict__ sk,
    const float* __restrict__ ow1,
    const float* __restrict__ ob1,
    const float* __restrict__ ow2,
    const float* __restrict__ ob2,
    float* __restrict__ out)
{
    __shared__ float g_lds[2][16 * 256];  // per-wave G matrix, 16 rows x 256 cols
    __shared__ float sk_t[2][256];        // sk tile  [b16][k16]
    __shared__ float w1_t[2][256];        // ow1 tile [k16][h16] (contiguous in global)
    __shared__ float b1_t[2][256];
    __shared__ float w2_t[2][256];

    const int wave = threadIdx.x >> 5;
    const int lane = threadIdx.x & 31;
    const int wgid = blockIdx.x * 2 + wave;          // 0..1023 tiles
    const int k_base = (wgid & 1) * 16;
    const int b_base = (wgid >> 1) * 16;

    // Cooperative per-wave tile loads
#pragma unroll
    for (int it = 0; it < 8; ++it) {
        const int i = it * 32 + lane;
        sk_t[wave][i] = sk[(b_base + (i >> 4)) * K_CLS + k_base + (i & 15)];
        w1_t[wave][i] = ow1[k_base * HID + i];
        b1_t[wave][i] = ob1[k_base * HID + i];
        w2_t[wave][i] = ow2[k_base * HID + i];
    }
    __syncthreads();

    // Build G = relu(sk*ow1 + ob1) in LDS: 4096 values / 32 lanes
#pragma unroll 4
    for (int t = 0; t < 128; ++t) {
        const int idx = t * 32 + lane;
        const int bb = idx >> 8;     // batch row in tile
        const int c  = idx & 255;    // (k',h) contraction index
        const int kp = c >> 4;
        const int h  = c & 15;
        const float v = fmaf(sk_t[wave][bb * 16 + kp],
                             w1_t[wave][kp * 16 + h],
                             b1_t[wave][kp * 16 + h]);
        g_lds[wave][bb * 256 + c] = fmaxf(v, 0.f);
    }
    __syncthreads();

    // 64 x V_WMMA_F32_16X16X4_F32, accumulating the block-diagonal GEMM.
    // A layout (16x4 f32): lanes 0-15 hold row M=lane, K={c0,c0+1};
    //                      lanes 16-31 hold row M=lane-16, K={c0+2,c0+3}.
    // B layout (4x16 f32): vgpr0 = rows K={0,2}, vgpr1 = rows K={1,3}; lane%16 = N.
    v8f acc = {};
    const int row = lane & 15;
    const int hi  = lane >> 4;   // 0 or 1 (half-wave)
#pragma unroll 4
    for (int cc = 0; cc < 64; ++cc) {
        const int c0    = cc * 4;
        const int kidx  = cc >> 2;        // k' block this chunk contributes to
        const int hbase = (cc & 3) * 4;

        // A fragment: two consecutive G columns -> aligned ds_load_b64
        v2f a = *(const v2f*)&g_lds[wave][row * 256 + c0 + hi * 2];

        // B fragment: only column n==kidx is nonzero (block-diagonal)
        const int h0 = hbase + hi * 2;
        const float s0 = w2_t[wave][kidx * 16 + h0];
        const float s1 = w2_t[wave][kidx * 16 + h0 + 1];
        const bool on = (row == kidx);
        v2f bv;
        bv.x = on ? s0 : 0.f;
        bv.y = on ? s1 : 0.f;

        acc = __builtin_amdgcn_wmma_f32_16x16x4_f32(
            /*neg_a=*/false, a, /*neg_b=*/false, bv,
            /*c_mod=*/(short)0, acc, /*reuse_a=*/false, /*reuse_b=*/false);
    }

    // Add ob2 and store C tile. C layout: vgpr r, lane l<16: (M=r, N=l);
    // l>=16: (M=r+8, N=l-16).
    const float bias = ob2[k_base + row];
#pragma unroll
    for (int r = 0; r < 8; ++r) {
        const int brow = r + hi * 8;
        out[(b_base + brow) * K_CLS + k_base + row] = acc[r] + bias;
    }
}

// ---------------------------------------------------------------------------
extern "C" void kernel_launch(void* const* d_in, const int* in_sizes, int n_in,
                              void* d_out, int out_size, void* d_ws, size_t ws_size,
                              hipStream_t stream) {
    const float* x   = (const float*)d_in[0];
    const float* iw1 = (const float*)d_in[1];
    const float* ib1 = (const float*)d_in[2];
    const float* iw2 = (const float*)d_in[3];
    const float* ib2 = (const float*)d_in[4];
    const float* ow1 = (const float*)d_in[5];
    const float* ob1 = (const float*)d_in[6];
    const float* ow2 = (const float*)d_in[7];
    const float* ob2 = (const float*)d_in[8];
    float* outp = (float*)d_out;

    float* sk = (float*)d_ws;  // B_TOT * K_CLS floats = 1 MB scratch

    dim3 gA(B_TOT / 256, K_CLS);
    kan_inner<<<gA, 256, 0, stream>>>(x, iw1, ib1, iw2, ib2, sk);

    // 1024 tiles of 16x16, 2 waves (tiles) per 64-thread block
    kan_outer<<<512, 64, 0, stream>>>(sk, ow1, ob1, ow2, ob2, outp);
}